// KANSE_38465727103293
// MI455X (gfx1250) — compile-verified
//
#include <hip/hip_runtime.h>
#include <math.h>

// ---------------------------------------------------------------------------
// KAN squeeze-excitation for MI455X (gfx1250, wave32).
//   A) channel mean over 64x64 planes      (memory bound, ~0.27 GB read)
//   B) 2 KAN linear layers as f32 WMMA GEMMs (single workgroup, LDS-staged)
//   C) x * sigmoid(gate)                   (memory bound, ~0.54 GB r+w)
// Roofline: ~0.81 GB HBM traffic @ 23.3 TB/s => ~35 us floor. Kernel B is
// kept to a few us: divisions replaced by uniform-grid reciprocals, spline
// bases computed once per (b,i) cooperatively into LDS (no 8x per-lane
// recompute), GEMMs on V_WMMA_F32_16X16X4_F32 (full f32 matrix path).
// ---------------------------------------------------------------------------

typedef __attribute__((ext_vector_type(2))) float v2f;
typedef __attribute__((ext_vector_type(4))) float v4f;
typedef __attribute__((ext_vector_type(8))) float v8f;

#define NB 32       // batch
#define NC 512      // channels (layer1 in / layer2 out)
#define NH 64       // hidden   (layer1 out / layer2 in)
#define PLANE 4096  // 64*64
#define CI 64       // layer-1 K-chunk (input features per LDS stage)

__device__ __forceinline__ float sigmoidf(float x) { return 1.0f / (1.0f + __expf(-x)); }
__device__ __forceinline__ float siluf(float x)    { return x * sigmoidf(x); }

// D = A(16x4) * B(4x16) + C, f32 WMMA (CDNA5)
__device__ __forceinline__ v8f wmma4(v2f a, v2f b, v8f c) {
    return __builtin_amdgcn_wmma_f32_16x16x4_f32(
        /*neg_a=*/false, a, /*neg_b=*/false, b,
        /*c_mod=*/(short)0, c, /*reuse_a=*/false, /*reuse_b=*/false);
}

// Cubic B-spline bases (GRID_SIZE=5, order=3): 12 knots -> 8 basis values.
// The knot grid from make_grid() is uniform (spacing h), so every Cox-de Boor
// denominator g[t+k]-g[t] == k*h: rk[k-1] holds its reciprocal (no divisions
// in the loop). Order-0 uses half-open intervals [g[t], g[t+1]).
__device__ __forceinline__ void bspline8(float x, const float (&g)[12],
                                         const float (&rk)[3], float (&out)[8]) {
    float b[11];
#pragma unroll
    for (int t = 0; t < 11; ++t)
        b[t] = (x >= g[t] && x < g[t + 1]) ? 1.0f : 0.0f;
#pragma unroll
    for (int k = 1; k <= 3; ++k) {
#pragma unroll
        for (int t = 0; t + k < 11; ++t)
            b[t] = rk[k - 1] * ((x - g[t]) * b[t] + (g[t + k + 1] - x) * b[t + 1]);
    }
#pragma unroll
    for (int t = 0; t < 8; ++t) out[t] = b[t];
}

// ---------------------------------------------------------------------------
// Kernel A: s[b,c] = nan_to_num(mean(x[b,c,:,:]))
// ---------------------------------------------------------------------------
__global__ void kanse_mean_kernel(const float* __restrict__ x, float* __restrict__ s) {
    const int bid = blockIdx.x;                 // b*512 + c
    const v4f* p = (const v4f*)(x + (size_t)bid * PLANE);
    const int t = threadIdx.x;
    float sum = 0.0f;
#pragma unroll
    for (int j = 0; j < 4; ++j) {
        v4f v = __builtin_nontemporal_load(&p[t + 256 * j]);
        sum += v.x + v.y + v.z + v.w;
    }
    __shared__ float red[256];
    red[t] = sum;
    __syncthreads();
#pragma unroll
    for (int off = 128; off > 0; off >>= 1) {
        if (t < off) red[t] += red[t + off];
        __syncthreads();
    }
    if (t == 0) {
        float m = red[0] * (1.0f / (float)PLANE);
        if (m != m) m = 0.0f;                   // nan_to_num
        s[bid] = m;
    }
}

// ---------------------------------------------------------------------------
// Kernel B: full KAN MLP in one workgroup (8 waves), f32 WMMA 16x16x4.
// A/B fragment layout (ISA 7.12.2, 32-bit 16x4): lane l, component v holds
// K = 4*step + 2*(l>>4) + v; M/N = l&15. C/D: M = 8*(l>>4)+r, N = l&15.
// ---------------------------------------------------------------------------
__global__ void __launch_bounds__(256)
kanse_kan_kernel(const float* __restrict__ s_buf,
                 const float* __restrict__ grid1,
                 const float* __restrict__ base_w1,
                 const float* __restrict__ spline_w1,
                 const float* __restrict__ scaler1,
                 const float* __restrict__ grid2,
                 const float* __restrict__ base_w2,
                 const float* __restrict__ spline_w2,
                 const float* __restrict__ scaler2,
                 float* __restrict__ gate_buf) {
    __shared__ float lds_sp[NB * CI * 8];       // spline-basis chunk  (64 KB)
    __shared__ float lds_sb[NB * CI];           // silu(input) chunk   ( 8 KB)
    __shared__ float h_lds[NB * NH];            // layer-2 input       ( 8 KB)

    const int wave = threadIdx.x >> 5;          // 0..7 (wave32)
    const int lane = threadIdx.x & 31;
    const int hi   = lane >> 4;                 // which K half this lane holds
    const int lo   = lane & 15;

    // All grid rows are identical by construction (make_grid tiles one row).
    float g1[12], g2[12];
#pragma unroll
    for (int t = 0; t < 12; ++t) { g1[t] = grid1[t]; g2[t] = grid2[t]; }
    float rk1[3], rk2[3];
#pragma unroll
    for (int k = 1; k <= 3; ++k) {
        rk1[k - 1] = 1.0f / (g1[k] - g1[0]);    // uniform grid: g[t+k]-g[t]==k*h
        rk2[k - 1] = 1.0f / (g2[k] - g2[0]);
    }

    // ------------------- layer 1: M=32 (2 tiles) x N=64 (4 tiles) ----------
    const int tm1 = wave >> 2, tn1 = wave & 3;  // one tile per wave
    const int bRow1 = tm1 * 16 + lo;
    const int oCol1 = tn1 * 16 + lo;
    v8f acc1 = {};

    for (int chunk = 0; chunk < NC / CI; ++chunk) {
        // stage: bases + silu for (b, i) pairs of this chunk (8 per thread)
        for (int p = threadIdx.x; p < NB * CI; p += 256) {
            const int b = p >> 6, il = p & (CI - 1);
            const float xv = s_buf[b * NC + chunk * CI + il];
            float bs[8];
            bspline8(xv, g1, rk1, bs);
            v4f* dst = (v4f*)&lds_sp[p * 8];
            dst[0] = (v4f){bs[0], bs[1], bs[2], bs[3]};
            dst[1] = (v4f){bs[4], bs[5], bs[6], bs[7]};
            lds_sb[p] = siluf(xv);
        }
        __syncthreads();

        // base_out partial: silu(s) @ base_w1^T over this chunk (K = 64)
        for (int step = 0; step < CI / 4; ++step) {
            const int k0 = step * 4 + hi * 2;
            v2f a = *(const v2f*)&lds_sb[bRow1 * CI + k0];
            v2f b = *(const v2f*)&base_w1[oCol1 * NC + chunk * CI + k0];
            acc1 = wmma4(a, b, acc1);
        }
        // spline_out partial: bases @ (spline_w1*scaler1)^T over this chunk
        for (int il = 0; il < CI; ++il) {
            const int i = chunk * CI + il;
            const float sc = scaler1[oCol1 * NC + i];
            const float* wp = spline_w1 + ((size_t)oCol1 * NC + i) * 8;
            const float* ap = &lds_sp[(bRow1 * CI + il) * 8];
            v2f a0 = *(const v2f*)&ap[2 * hi];        // K = 8i .. 8i+3
            v2f a1 = *(const v2f*)&ap[4 + 2 * hi];    // K = 8i+4 .. 8i+7
            v2f b0 = *(const v2f*)&wp[2 * hi];
            v2f b1 = *(const v2f*)&wp[4 + 2 * hi];
            b0 *= sc; b1 *= sc;
            acc1 = wmma4(a0, b0, acc1);
            acc1 = wmma4(a1, b1, acc1);
        }
        __syncthreads();                        // before next chunk overwrite
    }

    // h = silu(layer1 out)  (C/D layout: M = 8*hi + r, N = lo)
#pragma unroll
    for (int r = 0; r < 8; ++r) {
        const int b = tm1 * 16 + hi * 8 + r;
        h_lds[b * NH + oCol1] = siluf(acc1[r]);
    }
    __syncthreads();

    // ------------------- layer 2: M=32 (2) x N=512 (32) = 64 tiles ---------
    // stage: bases + silu for all (b, i) of layer-2 input (K = 64, one chunk)
    for (int p = threadIdx.x; p < NB * NH; p += 256) {
        const float xv = h_lds[p];
        float bs[8];
        bspline8(xv, g2, rk2, bs);
        v4f* dst = (v4f*)&lds_sp[p * 8];
        dst[0] = (v4f){bs[0], bs[1], bs[2], bs[3]};
        dst[1] = (v4f){bs[4], bs[5], bs[6], bs[7]};
        lds_sb[p] = siluf(xv);
    }
    __syncthreads();

    for (int tile = wave; tile < 64; tile += 8) {
        const int tm = tile >> 5, tn = tile & 31;
        const int bRow = tm * 16 + lo;
        const int oCol = tn * 16 + lo;
        v8f acc = {};

        // base_out: silu(h) @ base_w2^T, K = 64
        for (int step = 0; step < NH / 4; ++step) {
            const int k0 = step * 4 + hi * 2;
            v2f a = *(const v2f*)&lds_sb[bRow * NH + k0];
            v2f b = *(const v2f*)&base_w2[oCol * NH + k0];
            acc = wmma4(a, b, acc);
        }
        // spline_out: K = 64*8
        for (int i = 0; i < NH; ++i) {
            const float sc = scaler2[oCol * NH + i];
            const float* wp = spline_w2 + ((size_t)oCol * NH + i) * 8;
            const float* ap = &lds_sp[(bRow * NH + i) * 8];
            v2f a0 = *(const v2f*)&ap[2 * hi];
            v2f a1 = *(const v2f*)&ap[4 + 2 * hi];
            v2f b0 = *(const v2f*)&wp[2 * hi];
            v2f b1 = *(const v2f*)&wp[4 + 2 * hi];
            b0 *= sc; b1 *= sc;
            acc = wmma4(a0, b0, acc);
            acc = wmma4(a1, b1, acc);
        }
        // gate = sigmoid(layer2 out)
#pragma unroll
        for (int r = 0; r < 8; ++r) {
            const int b = tm * 16 + hi * 8 + r;
            gate_buf[b * NC + oCol] = sigmoidf(acc[r]);
        }
    }
}

// ---------------------------------------------------------------------------
// Kernel C: out = x * gate[b,c]   (streaming, non-temporal)
// ---------------------------------------------------------------------------
__global__ void kanse_gate_kernel(const float* __restrict__ x,
                                  const float* __restrict__ gate,
                                  float* __restrict__ out) {
    const int bid = blockIdx.x;                 // b*512 + c
    const float g = gate[bid];                  // scalar broadcast per block
    const v4f* p = (const v4f*)(x + (size_t)bid * PLANE);
    v4f* q = (v4f*)(out + (size_t)bid * PLANE);
    const int t = threadIdx.x;
#pragma unroll
    for (int j = 0; j < 4; ++j) {
        v4f v = __builtin_nontemporal_load(&p[t + 256 * j]);
        v *= g;
        __builtin_nontemporal_store(v, &q[t + 256 * j]);
    }
}

// ---------------------------------------------------------------------------
extern "C" void kernel_launch(void* const* d_in, const int* in_sizes, int n_in,
                              void* d_out, int out_size, void* d_ws, size_t ws_size,
                              hipStream_t stream) {
    const float* x         = (const float*)d_in[0];
    const float* grid1     = (const float*)d_in[1];
    const float* base_w1   = (const float*)d_in[2];
    const float* spline_w1 = (const float*)d_in[3];
    const float* scaler1   = (const float*)d_in[4];
    const float* grid2     = (const float*)d_in[5];
    const float* base_w2   = (const float*)d_in[6];
    const float* spline_w2 = (const float*)d_in[7];
    const float* scaler2   = (const float*)d_in[8];
    float* out = (float*)d_out;

    float* s_buf    = (float*)d_ws;             // 32*512 f32 = 64 KB
    float* gate_buf = s_buf + NB * NC;          // 32*512 f32 = 64 KB

    const int nplanes = NB * NC;                // 16384
    kanse_mean_kernel<<<nplanes, 256, 0, stream>>>(x, s_buf);
    kanse_kan_kernel<<<1, 256, 0, stream>>>(s_buf, grid1, base_w1, spline_w1, scaler1,
                                            grid2, base_w2, spline_w2, scaler2, gate_buf);
    kanse_gate_kernel<<<nplanes, 256, 0, stream>>>(x, gate_buf, out);
}